// GINMultiGraphModelwoSol_64742337020119
// MI455X (gfx1250) — compile-verified
//
#include <hip/hip_runtime.h>

typedef float v2f __attribute__((ext_vector_type(2)));
typedef float v8f __attribute__((ext_vector_type(8)));

// ---------------------------------------------------------------- zero fill
__global__ void fill_zero4_kernel(float4* __restrict__ p, long n4) {
  long i = (long)blockIdx.x * blockDim.x + threadIdx.x;
  if (i < n4) p[i] = make_float4(0.f, 0.f, 0.f, 0.f);
}

// ------------------------------------------------------- edge gather+scatter
// agg[dst[e]][f] += x[src[e]][f]; one thread per (edge, 4 features).
// 16 consecutive threads share an edge -> one coalesced 256B row gather via
// global_load_b128; the 51MB agg target is L2-resident (192MB L2) so the
// f32 atomics stay on-chip.
__global__ __launch_bounds__(256) void edge_scatter_kernel(
    const float4* __restrict__ x4, const int* __restrict__ ei,
    float* __restrict__ agg, int E) {
  long idx = (long)blockIdx.x * blockDim.x + threadIdx.x;
  if (idx >= (long)E * 16) return;
  int e  = (int)(idx >> 4);
  int f4 = (int)(idx & 15);
  int s = ei[e];        // src row
  int d = ei[E + e];    // dst row
  float4 v = x4[(long)s * 16 + f4];
  float* dst = &agg[(long)d * 64 + f4 * 4];
  atomicAdd(dst + 0, v.x);
  atomicAdd(dst + 1, v.y);
  atomicAdd(dst + 2, v.z);
  atomicAdd(dst + 3, v.w);
}

// ------------------------------------------------------------ fused GIN MLP
// Per wave: 16-node tile.  h = x+agg (16x64) staged in LDS;
// GEMM1 (W1, 64x64) via v_wmma_f32_16x16x4_f32, fused bias+BN+ReLU,
// GEMM2 (W2) + ReLU, write node output + atomic pooled segment-sum.
__global__ __launch_bounds__(128) void gin_mlp_kernel(
    const float* xin, const float* __restrict__ agg,
    float* xout,
    const float* __restrict__ W1, const float* __restrict__ b1,
    const float* __restrict__ gamma, const float* __restrict__ beta,
    const float* __restrict__ bmean, const float* __restrict__ bvar,
    const float* __restrict__ W2, const float* __restrict__ b2,
    const int* __restrict__ batch, float* __restrict__ pooled) {
  __shared__ float sW1[64 * 64];
  __shared__ float sW2[64 * 64];
  __shared__ float sH[4][16][64];   // per-wave 16x64 tile

  const int tid = threadIdx.x;
  const int wave = tid >> 5;
  const int lane = tid & 31;

  // cooperative weight staging (16KB + 16KB), b128 granularity
  {
    const float4* W14 = (const float4*)W1;
    const float4* W24 = (const float4*)W2;
    float4* sW14 = (float4*)sW1;
    float4* sW24 = (float4*)sW2;
    for (int i = tid; i < 1024; i += 128) {
      sW14[i] = W14[i];
      sW24[i] = W24[i];
    }
  }

  const int tile = blockIdx.x * 4 + wave;
  const long node0 = (long)tile * 16;

  // stage h = x + agg for this wave's 16 nodes (b128 loads + b128 LDS store)
  {
    const float4* xin4 = (const float4*)xin;
    const float4* agg4 = (const float4*)agg;
    for (int i = lane; i < 256; i += 32) {
      int r = i >> 4, c4 = i & 15;
      long g = (node0 + r) * 16 + c4;
      float4 a = xin4[g];
      float4 b = agg4[g];
      float4 h = make_float4(a.x + b.x, a.y + b.y, a.z + b.z, a.w + b.w);
      *(float4*)&sH[wave][r][c4 * 4] = h;
    }
  }
  __syncthreads();

  const int mrow = lane & 15;            // M (A) / N (B,C) sub-index
  const int koff = (lane < 16) ? 0 : 2;  // K half per ISA f32 16x4 layout
  const int mbase = (lane < 16) ? 0 : 8; // C-matrix M half

  // ---------------- GEMM1: (16x64) @ (64x64) with 16x16x4 f32 WMMA --------
  v8f acc[4] = {};
  for (int kk = 0; kk < 16; ++kk) {
    const int kb = kk * 4 + koff;
    v2f a;
    a.x = sH[wave][mrow][kb];
    a.y = sH[wave][mrow][kb + 1];
#pragma unroll
    for (int nt = 0; nt < 4; ++nt) {
      const int col = nt * 16 + mrow;
      v2f b;
      b.x = sW1[kb * 64 + col];
      b.y = sW1[(kb + 1) * 64 + col];
      acc[nt] = __builtin_amdgcn_wmma_f32_16x16x4_f32(
          false, a, false, b, (short)0, acc[nt], false, false);
    }
  }

  // fused bias + BatchNorm(eval) + ReLU, write back into the LDS tile
#pragma unroll
  for (int nt = 0; nt < 4; ++nt) {
    const int col = nt * 16 + mrow;
    const float bb = b1[col];
    const float rs = rsqrtf(bvar[col] + 1e-5f) * gamma[col];
    const float mu = bmean[col];
    const float be = beta[col];
#pragma unroll
    for (int r = 0; r < 8; ++r) {
      float v = acc[nt][r] + bb;
      v = (v - mu) * rs + be;
      sH[wave][mbase + r][col] = fmaxf(v, 0.0f);
    }
  }
  __syncthreads();

  // ---------------- GEMM2 ------------------------------------------------
  v8f acc2[4] = {};
  for (int kk = 0; kk < 16; ++kk) {
    const int kb = kk * 4 + koff;
    v2f a;
    a.x = sH[wave][mrow][kb];
    a.y = sH[wave][mrow][kb + 1];
#pragma unroll
    for (int nt = 0; nt < 4; ++nt) {
      const int col = nt * 16 + mrow;
      v2f b;
      b.x = sW2[kb * 64 + col];
      b.y = sW2[(kb + 1) * 64 + col];
      acc2[nt] = __builtin_amdgcn_wmma_f32_16x16x4_f32(
          false, a, false, b, (short)0, acc2[nt], false, false);
    }
  }

  // ReLU, node write (in place is safe: block-private rows) + pooled scatter
#pragma unroll
  for (int r = 0; r < 8; ++r) {
    const long node = node0 + mbase + r;
    const long gofs = (long)batch[node] * 64;
#pragma unroll
    for (int nt = 0; nt < 4; ++nt) {
      const int col = nt * 16 + mrow;
      const float v = fmaxf(acc2[nt][r] + b2[col], 0.0f);
      xout[node * 64 + col] = v;
      atomicAdd(&pooled[gofs + col], v);
    }
  }
}

// ------------------------------------------------------------- graph head
__global__ __launch_bounds__(64) void head_kernel(
    const float* __restrict__ pooled,  // [3][G][64]
    const int* __restrict__ acid_idx, const float* __restrict__ conc,
    const float* __restrict__ emb_acid, const float* __restrict__ emb_conc,
    const float* __restrict__ fc1W, const float* __restrict__ fc1b,
    const float* __restrict__ fc2W, const float* __restrict__ fc2b,
    float* __restrict__ out, int G) {
  __shared__ float feat[448];
  __shared__ float h[64];
  const int g = blockIdx.x;
  const int t = threadIdx.x;

  for (int i = t; i < 448; i += 64) {
    float v;
    if (i < 192) {
      int l = i >> 6, c = i & 63;
      v = pooled[((long)l * G + g) * 64 + c];
    } else if (i < 320) {
      v = emb_acid[acid_idx[g] * 128 + (i - 192)];
    } else {
      int ci = (conc[g] == 0.5f) ? 0 : 1;
      v = emb_conc[ci * 128 + (i - 320)];
    }
    feat[i] = v;
  }
  __syncthreads();

  float s = fc1b[t];
  for (int k = 0; k < 448; ++k) s += feat[k] * fc1W[k * 64 + t];
  h[t] = fmaxf(s, 0.0f);
  __syncthreads();

  if (t < 2) {
    float z = fc2b[t];
    for (int k = 0; k < 64; ++k) z += h[k] * fc2W[k * 2 + t];
    out[(long)g * 2 + t] = 1.0f / (1.0f + expf(-z));
  }
}

// ----------------------------------------------------------------- driver
extern "C" void kernel_launch(void* const* d_in, const int* in_sizes, int n_in,
                              void* d_out, int out_size, void* d_ws,
                              size_t ws_size, hipStream_t stream) {
  const float* x_a   = (const float*)d_in[0];
  const int*   ei    = (const int*)d_in[1];
  const int*   batch = (const int*)d_in[2];
  const int*   acid  = (const int*)d_in[3];
  const float* conc  = (const float*)d_in[4];
  const float* W1s   = (const float*)d_in[6];
  const float* b1s   = (const float*)d_in[7];
  const float* gam   = (const float*)d_in[8];
  const float* bet   = (const float*)d_in[9];
  const float* bmn   = (const float*)d_in[10];
  const float* bvr   = (const float*)d_in[11];
  const float* W2s   = (const float*)d_in[12];
  const float* b2s   = (const float*)d_in[13];
  const float* eA    = (const float*)d_in[14];
  const float* eC    = (const float*)d_in[15];
  const float* fc1W  = (const float*)d_in[16];
  const float* fc1b  = (const float*)d_in[17];
  const float* fc2W  = (const float*)d_in[18];
  const float* fc2b  = (const float*)d_in[19];

  const int N = in_sizes[0] / 64;  // 200000 (64 | N)
  const int E = in_sizes[1] / 2;   // 3200000
  const int G = in_sizes[3];       // 4096
  const long NF = (long)N * 64;

  float* agg    = (float*)d_ws;          // N*64
  float* xbuf   = agg + NF;              // N*64
  float* pooled = xbuf + NF;             // 3*G*64

  // zero pooled accumulators once per call
  {
    long n4 = 3L * G * 64 / 4;
    fill_zero4_kernel<<<(unsigned)((n4 + 255) / 256), 256, 0, stream>>>(
        (float4*)pooled, n4);
  }

  const unsigned scatterBlocks = (unsigned)(((long)E * 16 + 255) / 256);
  const unsigned mlpBlocks = (unsigned)(N / 64);  // 3125, exact
  const long NF4 = NF / 4;

  const float* xcur = x_a;
  for (int l = 0; l < 3; ++l) {
    fill_zero4_kernel<<<(unsigned)((NF4 + 255) / 256), 256, 0, stream>>>(
        (float4*)agg, NF4);
    edge_scatter_kernel<<<scatterBlocks, 256, 0, stream>>>(
        (const float4*)xcur, ei, agg, E);
    gin_mlp_kernel<<<mlpBlocks, 128, 0, stream>>>(
        xcur, agg, xbuf,
        W1s + (long)l * 64 * 64, b1s + l * 64,
        gam + l * 64, bet + l * 64, bmn + l * 64, bvr + l * 64,
        W2s + (long)l * 64 * 64, b2s + l * 64,
        batch, pooled + (long)l * G * 64);
    xcur = xbuf;  // layers 1,2 update xbuf in place (block-private rows)
  }

  head_kernel<<<(unsigned)G, 64, 0, stream>>>(
      pooled, acid, conc, eA, eC, fc1W, fc1b, fc2W, fc2b, (float*)d_out, G);
}